// SparsePool_11132555231810
// MI455X (gfx1250) — compile-verified
//
#include <hip/hip_runtime.h>
#include <hip/hip_bf16.h>
#include <math.h>

// Problem constants from the reference
#define OUT_G  32
#define NSEG   32768          // 32^3 output voxels
#define CCH    128            // channels
#define DEPTH  8              // async pipeline depth (rows in flight per wave)
#define WPB    8              // waves per block in the pool kernel

__device__ __forceinline__ int seg_of(const int* __restrict__ coords, int i) {
    int x = coords[3 * i + 0] >> 1;
    int y = coords[3 * i + 1] >> 1;
    int z = coords[3 * i + 2] >> 1;
    return (x * OUT_G + y) * OUT_G + z;
}

// ---------------- Phase 0: zero the histogram ----------------
__global__ void zero_cnt_kernel(int* __restrict__ cnt) {
    int i = blockIdx.x * blockDim.x + threadIdx.x;
    if (i < NSEG) cnt[i] = 0;
}

// ---------------- Phase 1: histogram points per segment ----------------
__global__ void count_kernel(const int* __restrict__ coords, int* __restrict__ cnt, int n) {
    int i = blockIdx.x * blockDim.x + threadIdx.x;
    int stride = gridDim.x * blockDim.x;
    for (; i < n; i += stride)
        atomicAdd(&cnt[seg_of(coords, i)], 1);
}

// ---------------- Phase 2: exclusive scan of 32768 counts (1 block) ----------------
__global__ __launch_bounds__(1024) void scan_kernel(const int* __restrict__ cnt,
                                                    int* __restrict__ offs,
                                                    int* __restrict__ cursor) {
    __shared__ int sh[1024];
    const int t = threadIdx.x;
    const int base = t * 32;
    int s = 0;
    #pragma unroll 8
    for (int j = 0; j < 32; ++j) s += cnt[base + j];
    sh[t] = s;
    __syncthreads();
    // inclusive Hillis-Steele scan over 1024 chunk sums
    for (int off = 1; off < 1024; off <<= 1) {
        int v = (t >= off) ? sh[t - off] : 0;
        __syncthreads();
        sh[t] += v;
        __syncthreads();
    }
    int run = (t == 0) ? 0 : sh[t - 1];     // exclusive prefix of this chunk
    for (int j = 0; j < 32; ++j) {
        offs[base + j]   = run;
        cursor[base + j] = run;
        run += cnt[base + j];
    }
    if (t == 1023) offs[NSEG] = run;        // total point count
}

// ---------------- Phase 3: scatter point indices into segment order ----------------
__global__ void scatter_kernel(const int* __restrict__ coords, int* __restrict__ cursor,
                               int* __restrict__ order, int n) {
    int i = blockIdx.x * blockDim.x + threadIdx.x;
    int stride = gridDim.x * blockDim.x;
    for (; i < n; i += stride) {
        int pos = atomicAdd(&cursor[seg_of(coords, i)], 1);
        order[pos] = i;
    }
}

// ---------------- Phase 4: gather-max, one wave per output voxel ----------------
// Each feature row (512B) is staged into LDS with ONE per-lane
// global_load_async_to_lds_b128 (lane L moves bytes [16L,16L+16)), DEPTH-deep
// pipelined via s_wait_asynccnt. Lane L accumulates channels 4L..4L+3.
__device__ __forceinline__ void issue_async_row(const float* __restrict__ feats,
                                                int row, int lane, uint32_t lds_slot_addr,
                                                bool first) {
    const uint64_t g = (uint64_t)(uintptr_t)(feats + (size_t)row * CCH + lane * 4);
    if (first) {
        asm volatile("global_load_async_to_lds_b128 %0, %1, off"
                     :: "v"(lds_slot_addr), "v"(g) : "memory");
    } else {
        // refill: make sure our ds_load of this slot retired before overwrite
        asm volatile("s_wait_dscnt 0\n\t"
                     "global_load_async_to_lds_b128 %0, %1, off"
                     :: "v"(lds_slot_addr), "v"(g) : "memory");
    }
}

__global__ __launch_bounds__(32 * WPB) void pool_kernel(const float* __restrict__ feats,
                                                        const int* __restrict__ order,
                                                        const int* __restrict__ offs,
                                                        float* __restrict__ out) {
    __shared__ __align__(16) float lbuf[WPB][DEPTH][CCH];   // 32 KB
    const int wave = threadIdx.x >> 5;
    const int lane = threadIdx.x & 31;
    const int s = blockIdx.x * WPB + wave;

    const int start = offs[s];
    const int n = offs[s + 1] - start;

    float4* outp = (float4*)(out + (size_t)s * CCH + lane * 4);
    if (n == 0) {                       // empty voxel -> zeros (wave-uniform branch)
        *outp = make_float4(0.f, 0.f, 0.f, 0.f);
        return;
    }

    // workgroup-relative LDS byte address for this wave/lane (low 32 bits of
    // the generic address of a __shared__ object are the LDS offset)
    const uint32_t lds_base = (uint32_t)(uintptr_t)(&lbuf[wave][0][0]) + (uint32_t)(lane * 16);

    // prologue: fill the pipeline
    const int P = (n < DEPTH) ? n : DEPTH;
    for (int k = 0; k < P; ++k)
        issue_async_row(feats, order[start + k], lane,
                        lds_base + (uint32_t)(k * (CCH * 4)), true);

    const float NEG_INF = -__builtin_inff();
    float4 acc = make_float4(NEG_INF, NEG_INF, NEG_INF, NEG_INF);

    // steady state: slack = n-1-i >= DEPTH-1  ->  constant wait
    const int mainN = n - (DEPTH - 1);          // may be <= 0
    int i = 0;
    for (; i < mainN; ++i) {
        asm volatile("s_wait_asynccnt 7" ::: "memory");
        const float4 v = *(const float4*)(&lbuf[wave][i & (DEPTH - 1)][lane * 4]);
        acc.x = fmaxf(acc.x, v.x);
        acc.y = fmaxf(acc.y, v.y);
        acc.z = fmaxf(acc.z, v.z);
        acc.w = fmaxf(acc.w, v.w);
        const int nx = i + DEPTH;
        if (nx < n)
            issue_async_row(feats, order[start + nx], lane,
                            lds_base + (uint32_t)((nx & (DEPTH - 1)) * (CCH * 4)), false);
    }

    // drain: slack = n-1-i in [DEPTH-2 .. 0], no more issues
    for (; i < n; ++i) {
        switch (n - 1 - i) {
        case 6: asm volatile("s_wait_asynccnt 6" ::: "memory"); break;
        case 5: asm volatile("s_wait_asynccnt 5" ::: "memory"); break;
        case 4: asm volatile("s_wait_asynccnt 4" ::: "memory"); break;
        case 3: asm volatile("s_wait_asynccnt 3" ::: "memory"); break;
        case 2: asm volatile("s_wait_asynccnt 2" ::: "memory"); break;
        case 1: asm volatile("s_wait_asynccnt 1" ::: "memory"); break;
        default: asm volatile("s_wait_asynccnt 0" ::: "memory"); break;
        }
        const float4 v = *(const float4*)(&lbuf[wave][i & (DEPTH - 1)][lane * 4]);
        acc.x = fmaxf(acc.x, v.x);
        acc.y = fmaxf(acc.y, v.y);
        acc.z = fmaxf(acc.z, v.z);
        acc.w = fmaxf(acc.w, v.w);
    }
    *outp = acc;
}

// ---------------- Fallback path (ws too small): atomic f32 max ----------------
__global__ void fill_neginf_kernel(float* __restrict__ out, int n4) {
    int i = blockIdx.x * blockDim.x + threadIdx.x;
    const float NEG_INF = -__builtin_inff();
    if (i < n4) ((float4*)out)[i] = make_float4(NEG_INF, NEG_INF, NEG_INF, NEG_INF);
}

__global__ __launch_bounds__(CCH) void atomic_pool_kernel(const float* __restrict__ feats,
                                                          const int* __restrict__ coords,
                                                          float* __restrict__ out, int n) {
    const int c = threadIdx.x;
    for (int p = blockIdx.x; p < n; p += gridDim.x) {
        const int sg = seg_of(coords, p);
        const float v = feats[(size_t)p * CCH + c];
        __hip_atomic_fetch_max(&out[(size_t)sg * CCH + c], v,
                               __ATOMIC_RELAXED, __HIP_MEMORY_SCOPE_AGENT);
    }
}

__global__ void fixup_kernel(float* __restrict__ out, int n) {
    int i = blockIdx.x * blockDim.x + threadIdx.x;
    const float NEG_INF = -__builtin_inff();
    if (i < n) {
        float v = out[i];
        out[i] = (v == NEG_INF) ? 0.f : v;
    }
}

extern "C" void kernel_launch(void* const* d_in, const int* in_sizes, int n_in,
                              void* d_out, int out_size, void* d_ws, size_t ws_size,
                              hipStream_t stream) {
    const float* feats = (const float*)d_in[0];
    const int* coords  = (const int*)d_in[1];
    const int N = in_sizes[1] / 3;
    float* out = (float*)d_out;

    // workspace layout (ints):
    //   cnt    [0, NSEG)
    //   offs   [NSEG, 2*NSEG]            (NSEG+1 entries)
    //   cursor [2*NSEG+64, 3*NSEG+64)
    //   order  [3*NSEG+128, 3*NSEG+128+N)
    int* ws     = (int*)d_ws;
    int* cnt    = ws;
    int* offs   = ws + NSEG;
    int* cursor = ws + 2 * NSEG + 64;
    int* order  = ws + 3 * NSEG + 128;
    const size_t need_bytes = ((size_t)(3 * NSEG + 128) + (size_t)N) * sizeof(int);

    if (d_ws != nullptr && ws_size >= need_bytes) {
        int blocks = (N + 255) / 256;
        if (blocks > 4096) blocks = 4096;
        zero_cnt_kernel<<<(NSEG + 255) / 256, 256, 0, stream>>>(cnt);
        count_kernel<<<blocks, 256, 0, stream>>>(coords, cnt, N);
        scan_kernel<<<1, 1024, 0, stream>>>(cnt, offs, cursor);
        scatter_kernel<<<blocks, 256, 0, stream>>>(coords, cursor, order, N);
        pool_kernel<<<NSEG / WPB, 32 * WPB, 0, stream>>>(feats, order, offs, out);
    } else {
        const int n4 = out_size / 4;
        fill_neginf_kernel<<<(n4 + 255) / 256, 256, 0, stream>>>(out, n4);
        atomic_pool_kernel<<<8192, CCH, 0, stream>>>(feats, coords, out, N);
        fixup_kernel<<<(out_size + 255) / 256, 256, 0, stream>>>(out, out_size);
    }
}